// SearchPredictModel_40621800685576
// MI455X (gfx1250) — compile-verified
//
#include <hip/hip_runtime.h>
#include <math.h>

// SearchPredictModel on MI455X (gfx1250, wave32).
// B=4096 L=200 F=8 D=64 K=15, E: (100001, 64) f32.
//
// Strategy: 1 block (256 thr = 8 wave32) per batch.
//  - Each wave processes 16-row chunks of e_all with V_WMMA_F32_16X16X4_F32:
//    A = 16 rows x 4 dims (16 steps cover D=64), B = self embedding broadcast
//    into all 16 columns -> D rows hold dot(e_all_row, e_self).
//  - Row norms from the same per-lane loads (partial sum-sq + shfl_xor(16)).
//  - sims to LDS, mask pos >= self_loc with -2, wave0 does K=15 iterative
//    argmax (index tie-break) + rank-sort of indices, then gather outputs.

typedef __attribute__((ext_vector_type(2))) float v2f;
typedef __attribute__((ext_vector_type(8))) float v8f;

#define B_      4096
#define L_      200
#define F_      8
#define D_      64
#define K_      15
#define NCHUNK  13      // ceil(200/16)

__global__ __launch_bounds__(256) void spm_topk_kernel(
    const int*   __restrict__ x,         // (B, L, F) int32
    const int*   __restrict__ self_loc,  // (B,) int32
    const float* __restrict__ E,         // (NFEAT+1, D) f32
    int*         __restrict__ out_x,     // (B, 16, F) int32 (bit pattern)
    float*       __restrict__ out_sim)   // (B, K) f32
{
    __shared__ float ldsDot[NCHUNK * 16];
    __shared__ float ldsN[NCHUNK * 16];
    __shared__ float ldsSim[L_];
    __shared__ float ldsSelfN;
    __shared__ int   topIdx[K_];
    __shared__ float topVal[K_];
    __shared__ int   sortedIdx[K_];
    __shared__ float sortedVal[K_];

    const int b    = blockIdx.x;
    const int tid  = threadIdx.x;
    const int lane = tid & 31;
    const int wave = tid >> 5;
    const int off  = (lane < 16) ? 0 : 2;   // K sub-offset per ISA A/B f32 layout
    const int m    = lane & 15;             // row within 16-row chunk

    const int sl       = self_loc[b];
    const int row_self = x[((size_t)b * L_ + sl) * F_ + (F_ - 3)];

    // ---- preload e_self in WMMA-B layout (per wave), plus its norm^2 ----
    const float* Es = E + (size_t)row_self * D_;
    float es0[16], es1[16];
    float npS = 0.f;
#pragma unroll
    for (int s = 0; s < 16; ++s) {
        float2 v = *(const float2*)(Es + 4 * s + off);
        es0[s] = v.x; es1[s] = v.y;
        npS += v.x * v.x + v.y * v.y;
    }
    const float nSelf = npS + __shfl_xor(npS, 16);
    if (tid == 0) ldsSelfN = nSelf;

    // ---- per-wave chunks: dots via WMMA f32 16x16x4, norms via VALU ----
    for (int c = wave; c < NCHUNK; c += 8) {
        const int l  = c * 16 + m;
        const int lc = (l < L_) ? l : (L_ - 1);     // clamp pad rows
        const int row = x[((size_t)b * L_ + lc) * F_ + (F_ - 3)];
        const float* Er = E + (size_t)row * D_;

        float a0[16], a1[16];
        float np = 0.f;
#pragma unroll
        for (int s = 0; s < 16; ++s) {
            float2 v = *(const float2*)(Er + 4 * s + off);
            a0[s] = v.x; a1[s] = v.y;
            np += v.x * v.x + v.y * v.y;
        }

        v8f acc = {0.f, 0.f, 0.f, 0.f, 0.f, 0.f, 0.f, 0.f};
#pragma unroll
        for (int s = 0; s < 16; ++s) {
            v2f A;  A[0]  = a0[s];  A[1]  = a1[s];
            v2f Bv; Bv[0] = es0[s]; Bv[1] = es1[s];
            // (neg_a, A, neg_b, B, c_mod, C, reuse_a, reuse_b)
            acc = __builtin_amdgcn_wmma_f32_16x16x4_f32(
                false, A, false, Bv, (short)0, acc, false, false);
        }

        const float nfull = np + __shfl_xor(np, 16);
        if (lane < 16 && l < L_) ldsN[l] = nfull;

        // D layout: v0..7 = rows 0..7 (lanes 0-15) / rows 8..15 (lanes 16-31),
        // columns identical -> extract from lane 0 and lane 16.
        if (lane == 0) {
#pragma unroll
            for (int j = 0; j < 8; ++j) {
                int l0 = c * 16 + j;
                if (l0 < L_) ldsDot[l0] = acc[j];
            }
        }
        if (lane == 16) {
#pragma unroll
            for (int j = 0; j < 8; ++j) {
                int l0 = c * 16 + 8 + j;
                if (l0 < L_) ldsDot[l0] = acc[j];
            }
        }
    }
    __syncthreads();

    // ---- cosine + validity mask ----
    if (tid < L_) {
        const float dot = ldsDot[tid];
        const float n2  = ldsN[tid];
        const float s1  = sqrtf(ldsSelfN + 1e-8f);
        const float s2  = sqrtf(n2 + 1e-8f);
        float sim = dot / (s1 * s2);
        if (tid >= sl) sim = -2.0f;      // valid = pos < self_loc (strict)
        ldsSim[tid] = sim;
    }
    __syncthreads();

    // ---- top-K by wave 0: iterative wave argmax, tie-break = smaller index ----
    if (tid < 32) {
        for (int k = 0; k < K_; ++k) {
            float best = -1e30f;
            int   bi   = L_;
            for (int l = lane; l < L_; l += 32) {
                float v = ldsSim[l];
                if (v > best) { best = v; bi = l; }   // l ascending => first occurrence
            }
#pragma unroll
            for (int o = 16; o > 0; o >>= 1) {
                float ov = __shfl_xor(best, o);
                int   oi = __shfl_xor(bi, o);
                if (ov > best || (ov == best && oi < bi)) { best = ov; bi = oi; }
            }
            // every lane holds the same (best, bi): write redundantly to avoid
            // cross-lane LDS ordering concerns inside the wave.
            topIdx[k] = bi;
            topVal[k] = best;
            ldsSim[bi] = -3.0f;          // knock out (below any valid/masked sim)
        }
        // rank-sort the K distinct indices ascending (parallel rank)
        if (lane < K_) {
            const int my = topIdx[lane];
            int r = 0;
#pragma unroll
            for (int j = 0; j < K_; ++j) r += (topIdx[j] < my) ? 1 : 0;
            sortedIdx[r] = my;
            sortedVal[r] = topVal[lane];
        }
    }
    __syncthreads();

    // ---- outputs: x_out (B,16,F) int32, sim_topk (B,K) f32 ----
    if (tid < 16 * F_) {
        const int r = tid >> 3;
        const int f = tid & 7;
        const int l = (r < K_) ? sortedIdx[r] : sl;   // row 15 = self row
        out_x[((size_t)b * 16 + r) * F_ + f] = x[((size_t)b * L_ + l) * F_ + f];
    }
    if (tid >= 128 && tid < 128 + K_) {
        out_sim[(size_t)b * K_ + (tid - 128)] = sortedVal[tid - 128];
    }
}

extern "C" void kernel_launch(void* const* d_in, const int* in_sizes, int n_in,
                              void* d_out, int out_size, void* d_ws, size_t ws_size,
                              hipStream_t stream) {
    (void)in_sizes; (void)n_in; (void)d_ws; (void)ws_size; (void)out_size;
    const int*   x        = (const int*)d_in[0];     // (B, L, F) int32
    const int*   self_loc = (const int*)d_in[1];     // (B,) int32
    const float* E        = (const float*)d_in[2];   // (NFEAT+1, D) f32

    // Output tuple flattened: x_out (int32, B*16*F elements) then sim (f32, B*K).
    int*   out_x   = (int*)d_out;
    float* out_sim = (float*)d_out + (size_t)B_ * 16 * F_;

    spm_topk_kernel<<<B_, 256, 0, stream>>>(x, self_loc, E, out_x, out_sim);
}